// LambdaLayerV2_72378788873001
// MI455X (gfx1250) — compile-verified
//
#include <hip/hip_runtime.h>

// ---------------------------------------------------------------------------
// LambdaLayerV2 on MI455X (gfx1250), fp32 end-to-end via V_WMMA_F32_16X16X4_F32.
//
// Pipeline (all on `stream`):
//   memset: zero S / denom / colmax accumulators
//   K1: key = context @ Wk          (WMMA, store key ws, atomic col-max)
//   K2: P = exp(key-max) in LDS; S += P^T @ context  (WMMA + f32 atomics);
//       denom += column sums of P
//   K3: lam = (S @ Wv)/denom; BN over (B,OUT) per mid; Weff[b] = Wq @ bn[b]
//   K4: out[b] = feature[b] @ Weff[b]                (WMMA, bulk pass)
// ---------------------------------------------------------------------------

#define B_    8
#define NQ_   16384
#define NK_   16384
#define FEAT_ 128
#define MID_  16
#define CTX_  128
#define OUT_  128
#define EPS_  1e-5f

typedef float v2f __attribute__((ext_vector_type(2)));
typedef float v8f __attribute__((ext_vector_type(8)));

static __device__ __forceinline__ v8f wmma4(v2f a, v2f b, v8f c) {
  // D(16x16,f32) = A(16x4,f32) x B(4x16,f32) + C
  return __builtin_amdgcn_wmma_f32_16x16x4_f32(false, a, false, b, (short)0, c,
                                               false, false);
}

// order-preserving float<->uint map for atomicMax on floats (any sign)
static __device__ __forceinline__ unsigned fenc(float f) {
  unsigned u = __float_as_uint(f);
  return (u & 0x80000000u) ? ~u : (u | 0x80000000u);
}
static __device__ __forceinline__ float fdec(unsigned e) {
  unsigned u = (e & 0x80000000u) ? (e & 0x7fffffffu) : ~e;
  return __uint_as_float(u);
}

// ---------------------------------------------------------------------------
// K1: key[b,n,m] = context[b,n,:] @ Wk   (+ per-(b,m) running max via atomics)
// block = 256 thr (8 waves); each wave does one 16-row strip, tile K=128.
// ---------------------------------------------------------------------------
__global__ __launch_bounds__(256) void k1_keyproj(
    const float* __restrict__ ctx, const float* __restrict__ Wk,
    float* __restrict__ keyw, unsigned* __restrict__ colmax) {
  const int lane = threadIdx.x & 31;
  const int wv   = threadIdx.x >> 5;
  const int n    = lane & 15;   // column (mid) for B/C/D; row (m) for A
  const int kh   = lane >> 4;   // k-half select
  const int row0 = blockIdx.x * 128 + wv * 16;       // flat row in [B*NK)
  const int b    = row0 / NK_;

  const float* __restrict__ arow = ctx + (size_t)(row0 + n) * FEAT_;
  v8f acc = {0.f, 0.f, 0.f, 0.f, 0.f, 0.f, 0.f, 0.f};
#pragma unroll
  for (int kb = 0; kb < FEAT_ / 4; ++kb) {
    const int k = kb * 4 + kh * 2;
    v2f a;  a[0] = arow[k];            a[1] = arow[k + 1];
    v2f bb; bb[0] = Wk[k * MID_ + n];  bb[1] = Wk[(k + 1) * MID_ + n];
    acc = wmma4(a, bb, acc);
  }

  float mx = acc[0];
#pragma unroll
  for (int r = 0; r < 8; ++r) {
    keyw[(size_t)(row0 + r + 8 * kh) * MID_ + n] = acc[r];
    mx = fmaxf(mx, acc[r]);
  }
  mx = fmaxf(mx, __shfl_xor(mx, 16, 32));   // combine the two row-halves
  if (kh == 0) atomicMax(&colmax[b * MID_ + n], fenc(mx));
}

// ---------------------------------------------------------------------------
// K2: S[b,m,c] += sum_n exp(key[n,m]-max[m]) * ctx[n,c] ; denom[b,m] += col sums
// 4 tiles of 128 rows per block (K accumulates to 512 before the atomics).
// P tile lives in LDS; context B-fragments stream straight from global.
// ---------------------------------------------------------------------------
#define K2_TILES 4
__global__ __launch_bounds__(256) void k2_lambda_acc(
    const float* __restrict__ ctx, const float* __restrict__ keyw,
    const unsigned* __restrict__ colmax, float* __restrict__ Sacc,
    float* __restrict__ denom) {
  __shared__ float P[128][MID_];      // 8 KB
  __shared__ float cmaxS[MID_];

  const int t    = threadIdx.x;
  const int lane = t & 31;
  const int wv   = t >> 5;            // wave == output c-tile (8 of them)
  const int n    = lane & 15;
  const int kh   = lane >> 4;
  const int rowB = blockIdx.x * (128 * K2_TILES);
  const int b    = rowB / NK_;

  if (t < MID_) cmaxS[t] = fdec(colmax[b * MID_ + t]);
  __syncthreads();

  v8f acc = {0.f, 0.f, 0.f, 0.f, 0.f, 0.f, 0.f, 0.f};
  float dsum = 0.f;

  for (int tt = 0; tt < K2_TILES; ++tt) {
    const int row0 = rowB + tt * 128;
    // P = exp(key - max): thread t handles row t>>1, 8 cols.
    {
      const int r  = t >> 1;
      const int m0 = (t & 1) * 8;
      const float* kr = keyw + (size_t)(row0 + r) * MID_ + m0;
#pragma unroll
      for (int i = 0; i < 8; ++i)
        P[r][m0 + i] = __expf(kr[i] - cmaxS[m0 + i]);
    }
    __syncthreads();

    if (t < MID_) {                       // denom partials (wave 0, lanes 0-15)
      float s = 0.f;
      for (int r = 0; r < 128; ++r) s += P[r][t];
      dsum += s;
    }

    // S_tile += P^T (16 x 128) @ C_tile (128 x 128); this wave's 16-col slice
#pragma unroll
    for (int kb = 0; kb < 32; ++kb) {
      const int kr = kb * 4 + kh * 2;
      v2f a;  a[0] = P[kr][n];  a[1] = P[kr + 1][n];
      const float* brow = ctx + (size_t)(row0 + kr) * CTX_ + wv * 16 + n;
      v2f bb; bb[0] = brow[0];  bb[1] = brow[CTX_];
      acc = wmma4(a, bb, acc);
    }
    __syncthreads();                      // before P is overwritten
  }

#pragma unroll
  for (int r = 0; r < 8; ++r)
    atomicAdd(&Sacc[(size_t)b * MID_ * CTX_ + (r + 8 * kh) * CTX_ + wv * 16 + n],
              acc[r]);
  if (t < MID_) atomicAdd(&denom[b * MID_ + t], dsum);
}

// ---------------------------------------------------------------------------
// K3 (single block): lam = (S @ Wv)/denom -> BN(train) -> Weff[b] = Wq @ bn[b]
// lam kept in a small global scratch (16K floats); WMMA for both matmuls.
// ---------------------------------------------------------------------------
__global__ __launch_bounds__(256) void k3_bn_weff(
    const float* __restrict__ Sacc, const float* __restrict__ denom,
    const float* __restrict__ Wv, const float* __restrict__ Wq,
    const float* __restrict__ gamma, const float* __restrict__ beta,
    float* __restrict__ lamG, float* __restrict__ Weff) {
  __shared__ float red1[128], red2[128], scl[MID_], shf[MID_];

  const int t    = threadIdx.x;
  const int lane = t & 31;
  const int wv   = t >> 5;
  const int n    = lane & 15;
  const int kh   = lane >> 4;

  // phase 1: lam rows (flat b*16+m = 128 rows) x 128 cols; wave wv = row strip
  for (int tc = 0; tc < 8; ++tc) {
    v8f acc = {0.f, 0.f, 0.f, 0.f, 0.f, 0.f, 0.f, 0.f};
#pragma unroll
    for (int kb = 0; kb < 32; ++kb) {
      const int k = kb * 4 + kh * 2;
      const float* ar = Sacc + (size_t)(wv * 16 + n) * CTX_ + k;
      v2f a;  a[0] = ar[0];                       a[1] = ar[1];
      v2f bb; bb[0] = Wv[k * OUT_ + tc * 16 + n]; bb[1] = Wv[(k + 1) * OUT_ + tc * 16 + n];
      acc = wmma4(a, bb, acc);
    }
#pragma unroll
    for (int r = 0; r < 8; ++r) {
      const int row = wv * 16 + r + 8 * kh;       // = b*16 + m
      lamG[row * OUT_ + tc * 16 + n] = acc[r] / denom[row];
    }
  }
  __threadfence_block();
  __syncthreads();

  // phase 2: BN stats over (B, OUT) per mid channel m (biased variance)
  if (t < 128) {
    const int m = t & 15, p = t >> 4, row = p * 16 + m;
    float s1 = 0.f, s2 = 0.f;
    for (int d = 0; d < OUT_; ++d) {
      const float v = lamG[row * OUT_ + d];
      s1 += v; s2 += v * v;
    }
    red1[t] = s1; red2[t] = s2;
  }
  __syncthreads();
  if (t < MID_) {
    float s1 = 0.f, s2 = 0.f;
    for (int p = 0; p < 8; ++p) { s1 += red1[p * 16 + t]; s2 += red2[p * 16 + t]; }
    const float inv  = 1.f / (float)(B_ * OUT_);
    const float mean = s1 * inv;
    const float var  = s2 * inv - mean * mean;
    const float sc   = gamma[t] * rsqrtf(var + EPS_);
    scl[t] = sc; shf[t] = beta[t] - mean * sc;
  }
  __syncthreads();

  // phase 3: bn in place
  for (int i = t; i < 128 * OUT_; i += 256) {
    const int row = i >> 7, d = i & 127, m = row & 15;
    lamG[row * OUT_ + d] = lamG[row * OUT_ + d] * scl[m] + shf[m];
  }
  __threadfence_block();
  __syncthreads();

  // phase 4: Weff[b] = Wq (128x16) @ bn[b] (16x128); 512 16x16 tiles, K=16
  for (int tid = wv; tid < 512; tid += 8) {
    const int b = tid >> 6, sub = tid & 63, tr = sub >> 3, tc = sub & 7;
    v8f acc = {0.f, 0.f, 0.f, 0.f, 0.f, 0.f, 0.f, 0.f};
#pragma unroll
    for (int kb = 0; kb < 4; ++kb) {
      const int k = kb * 4 + kh * 2;
      v2f a;  a[0] = Wq[(tr * 16 + n) * MID_ + k];
      a[1] = Wq[(tr * 16 + n) * MID_ + k + 1];
      v2f bb; bb[0] = lamG[(b * 16 + k) * OUT_ + tc * 16 + n];
      bb[1] = lamG[(b * 16 + k + 1) * OUT_ + tc * 16 + n];
      acc = wmma4(a, bb, acc);
    }
#pragma unroll
    for (int r = 0; r < 8; ++r)
      Weff[(size_t)b * FEAT_ * OUT_ + (tr * 16 + r + 8 * kh) * OUT_ + tc * 16 + n] = acc[r];
  }
}

// ---------------------------------------------------------------------------
// K4: out[b] = feature[b] (16384x128) @ Weff[b] (128x128). The bulk pass.
// A-fragments hoisted to registers (reused across all 8 column tiles).
// ---------------------------------------------------------------------------
__global__ __launch_bounds__(256) void k4_out(
    const float* __restrict__ feat, const float* __restrict__ Weff,
    float* __restrict__ out) {
  const int lane = threadIdx.x & 31;
  const int wv   = threadIdx.x >> 5;
  const int n    = lane & 15;
  const int kh   = lane >> 4;
  const int row0 = blockIdx.x * 128 + wv * 16;        // flat row in [B*NQ)
  const int b    = row0 / NQ_;
  const float* __restrict__ wb = Weff + (size_t)b * FEAT_ * OUT_;
  const float* __restrict__ ar = feat + (size_t)(row0 + n) * FEAT_;

  v2f afr[32];
#pragma unroll
  for (int kb = 0; kb < 32; ++kb) {
    const int k = kb * 4 + kh * 2;
    afr[kb][0] = ar[k]; afr[kb][1] = ar[k + 1];
  }

  for (int tc = 0; tc < 8; ++tc) {
    v8f acc = {0.f, 0.f, 0.f, 0.f, 0.f, 0.f, 0.f, 0.f};
#pragma unroll
    for (int kb = 0; kb < 32; ++kb) {
      const int k = kb * 4 + kh * 2;
      v2f bb; bb[0] = wb[k * OUT_ + tc * 16 + n];
      bb[1] = wb[(k + 1) * OUT_ + tc * 16 + n];
      acc = wmma4(afr[kb], bb, acc);
    }
#pragma unroll
    for (int r = 0; r < 8; ++r)
      out[(size_t)(row0 + r + 8 * kh) * OUT_ + tc * 16 + n] = acc[r];
  }
}

// ---------------------------------------------------------------------------
extern "C" void kernel_launch(void* const* d_in, const int* in_sizes, int n_in,
                              void* d_out, int out_size, void* d_ws, size_t ws_size,
                              hipStream_t stream) {
  (void)in_sizes; (void)n_in; (void)out_size; (void)ws_size;
  const float* feature = (const float*)d_in[0];
  const float* context = (const float*)d_in[1];
  const float* Wq      = (const float*)d_in[2];
  const float* Wk      = (const float*)d_in[3];
  const float* Wv      = (const float*)d_in[4];
  const float* gamma   = (const float*)d_in[5];
  const float* beta    = (const float*)d_in[6];
  float* out = (float*)d_out;

  char* ws = (char*)d_ws;
  float*    keyw   = (float*)ws;                       // B*NK*MID
  size_t off = (size_t)B_ * NK_ * MID_ * sizeof(float);
  float*    Sacc   = (float*)(ws + off); off += (size_t)B_ * MID_ * CTX_ * sizeof(float);
  float*    denom  = (float*)(ws + off); off += (size_t)B_ * MID_ * sizeof(float);
  unsigned* colmax = (unsigned*)(ws + off); off += (size_t)B_ * MID_ * sizeof(unsigned);
  float*    Weff   = (float*)(ws + off); off += (size_t)B_ * FEAT_ * OUT_ * sizeof(float);
  float*    lamG   = (float*)(ws + off);               // 128*OUT floats

  // zero the atomic accumulators (Sacc | denom | colmax are contiguous)
  hipMemsetAsync(Sacc, 0,
                 ((size_t)B_ * MID_ * CTX_ + 2 * (size_t)B_ * MID_) * sizeof(float),
                 stream);

  k1_keyproj   <<<(B_ * NK_) / 128,              256, 0, stream>>>(context, Wk, keyw, colmax);
  k2_lambda_acc<<<(B_ * NK_) / (128 * K2_TILES), 256, 0, stream>>>(context, keyw, colmax, Sacc, denom);
  k3_bn_weff   <<<1,                             256, 0, stream>>>(Sacc, denom, Wv, Wq, gamma, beta, lamG, Weff);
  k4_out       <<<(B_ * NQ_) / 128,              256, 0, stream>>>(feature, Weff, out);
}